// EmbeddingDropout_46918222741585
// MI455X (gfx1250) — compile-verified
//
#include <hip/hip_runtime.h>
#include <hip/hip_bf16.h>

// ---------------------------------------------------------------------------
// EmbeddingDropout: out[row, :] = keep[words[row]] * W[words[row], :]
// keep[v] = (threefry_uniform(key=42, v) >= 0.1) ? 1/0.9 : 0
// Memory-bound gather: no matmul content -> no WMMA. Optimize the data path:
// wave32, b128 loads, non-temporal b128 stores (keep 192MB L2 for the table).
// ---------------------------------------------------------------------------

typedef __attribute__((ext_vector_type(4))) float v4f;

__device__ __forceinline__ unsigned rotl32(unsigned x, int r) {
    return (x << r) | (x >> (32 - r));
}

// JAX threefry2x32 block cipher (matches jax/_src/prng.py exactly).
__device__ __forceinline__ void threefry2x32(unsigned ks0, unsigned ks1,
                                             unsigned x0, unsigned x1,
                                             unsigned& o0, unsigned& o1) {
    const unsigned ks2 = ks0 ^ ks1 ^ 0x1BD11BDAu;
    const int ra[4] = {13, 15, 26, 6};
    const int rb[4] = {17, 29, 16, 24};
    x0 += ks0; x1 += ks1;
#pragma unroll
    for (int i = 0; i < 4; ++i) { x0 += x1; x1 = rotl32(x1, ra[i]); x1 ^= x0; }
    x0 += ks1; x1 += ks2 + 1u;
#pragma unroll
    for (int i = 0; i < 4; ++i) { x0 += x1; x1 = rotl32(x1, rb[i]); x1 ^= x0; }
    x0 += ks2; x1 += ks0 + 2u;
#pragma unroll
    for (int i = 0; i < 4; ++i) { x0 += x1; x1 = rotl32(x1, ra[i]); x1 ^= x0; }
    x0 += ks0; x1 += ks1 + 3u;
#pragma unroll
    for (int i = 0; i < 4; ++i) { x0 += x1; x1 = rotl32(x1, rb[i]); x1 ^= x0; }
    x0 += ks1; x1 += ks2 + 4u;
#pragma unroll
    for (int i = 0; i < 4; ++i) { x0 += x1; x1 = rotl32(x1, ra[i]); x1 ^= x0; }
    x0 += ks2; x1 += ks0 + 5u;
    o0 = x0; o1 = x1;
}

// Kernel 1: keep[v] per vocab row, reproducing
//   keep = (jax.random.uniform(key(42), (V,1)) >= 0.1) / 0.9
// jax.random.key(42) -> threefry key words (0, 42).
// threefry_2x32 over iota(V): odd V padded to 2*half, split into
// (x0 = v, x1 = v+half) lanes; out = concat(out0_lanes, out1_lanes)[:V].
__global__ void __launch_bounds__(256)
embdrop_keep_kernel(float* __restrict__ keep, int V, int half) {
    int v = blockIdx.x * blockDim.x + threadIdx.x;
    if (v >= V) return;

    unsigned c0, c1;
    if (v < half) { c0 = (unsigned)v;          c1 = (unsigned)(v + half); }
    else          { c0 = (unsigned)(v - half); c1 = (unsigned)v;          }

    unsigned o0, o1;
    threefry2x32(0u, 42u, c0, c1, o0, o1);
    unsigned bits = (v < half) ? o0 : o1;

    // bits -> uniform [0,1): bitcast((bits>>9)|0x3f800000) - 1.0, clamped >= 0
    unsigned fb = (bits >> 9) | 0x3f800000u;
    float u = __uint_as_float(fb) - 1.0f;
    u = fmaxf(u, 0.0f);

    keep[v] = (u >= 0.1f) ? (1.0f / 0.9f) : 0.0f;
}

// Kernel 2: one 256-thread (8-wave) block per 4KB output row.
// Reads W row with regular (RT) b128 loads -> table stays hot in 192MB L2.
// Writes output with non-temporal b128 stores -> streaming, no L2 pollution.
__global__ void __launch_bounds__(256)
embdrop_gather_kernel(const int* __restrict__ words,
                      const float* __restrict__ W,
                      const float* __restrict__ keep,
                      float* __restrict__ out,
                      int nrows, int D4) {
    int row = blockIdx.x;
    if (row >= nrows) return;

    int idx = words[row];             // uniform across block -> scalar load
    float k = keep[idx];              // L2-resident 200KB LUT

    const int D = D4 * 4;
    const v4f* __restrict__ src = (const v4f*)(W   + (size_t)idx * (size_t)D);
    v4f*       __restrict__ dst = (v4f*)      (out + (size_t)row * (size_t)D);

    for (int j = threadIdx.x; j < D4; j += 256) {
        v4f v = src[j];               // global_load_b128 (RT)
        v *= k;
        __builtin_nontemporal_store(v, dst + j);  // global_store_b128 nt
    }
}

extern "C" void kernel_launch(void* const* d_in, const int* in_sizes, int n_in,
                              void* d_out, int out_size, void* d_ws, size_t ws_size,
                              hipStream_t stream) {
    const int*   words = (const int*)d_in[0];    // [B*S] int32 indices
    const float* W     = (const float*)d_in[1];  // [V*D] float32 table
    float*       out   = (float*)d_out;          // [B*S*D] float32
    float*       keep  = (float*)d_ws;           // V floats of scratch (~200KB)

    const int V     = 50257;
    const int nrows = in_sizes[0];               // B*S = 65536
    const int D     = in_sizes[1] / V;           // 1024
    const int D4    = D / 4;
    const int half  = (V + 1) / 2;               // threefry odd-size split

    embdrop_keep_kernel<<<(V + 255) / 256, 256, 0, stream>>>(keep, V, half);
    embdrop_gather_kernel<<<nrows, 256, 0, stream>>>(words, W, keep, out,
                                                     nrows, D4);
}